// ZIMoL_Loss_86955907875162
// MI455X (gfx1250) — compile-verified
//
#include <hip/hip_runtime.h>
#include <hip/hip_bf16.h>

// ZIMoL loss: fused BCE + zero-inflated mixture-of-logistics likelihood, full
// global reduction to one scalar.  Bandwidth-bound (184.5 MB read once ->
// ~7.9us at 23.3 TB/s).  Strategy: streaming B128 NT loads, hardware
// transcendentals (v_exp/v_log/v_rcp), deterministic two-stage reduction with
// the final cross-block stage done on the matrix unit (v_wmma_f32_16x16x4_f32
// with an all-ones A matrix).

typedef __attribute__((ext_vector_type(4))) float v4f;
typedef __attribute__((ext_vector_type(2))) float v2f;
typedef __attribute__((ext_vector_type(8))) float v8f;

#define ZIMOL_NBLOCKS 2048   // multiple of 64 (finalize WMMA chunk size)
#define ZIMOL_THREADS 256    // 8 wave32 per block
#define ZIMOL_EPS     1e-8f
#define ZIMOL_HB      0.05f  // HALF_BIN

__device__ __forceinline__ float fast_sigmoid(float x) {
    // 1 / (1 + e^-x) via v_exp_f32 + v_rcp_f32
    return __builtin_amdgcn_rcpf(1.0f + __expf(-x));
}

__device__ __forceinline__ void zimol_element(float p, float tvv,
                                              const float* __restrict__ mu,
                                              const float* __restrict__ sg,
                                              const float* __restrict__ wt,
                                              long long idx, long long B, int K,
                                              float& acc_bce, float& acc_ll,
                                              float& acc_np) {
    const bool paid = tvv > 0.0f;
    const float logp   = fmaxf(__logf(p), -100.0f);
    const float log1mp = fmaxf(__logf(1.0f - p), -100.0f);
    acc_bce += paid ? logp : log1mp;

    float lik = 0.0f;
    for (int k = 0; k < K; ++k) {
        const long long o = (long long)k * B + idx;
        const float m = mu[o];
        const float s = sg[o];
        const float w = wt[o];
        const float inv = __builtin_amdgcn_rcpf(s + ZIMOL_EPS);
        const float cu = fast_sigmoid((tvv + ZIMOL_HB - m) * inv);
        const float cl = fast_sigmoid((tvv - ZIMOL_HB - m) * inv);
        lik += w * fmaxf(cu - cl, ZIMOL_EPS);
    }
    const float ll = __logf(lik + ZIMOL_EPS);
    acc_ll += paid ? ll : 0.0f;
    acc_np += paid ? 1.0f : 0.0f;
}

__global__ void __launch_bounds__(ZIMOL_THREADS)
zimol_main_kernel(const float* __restrict__ p, const float* __restrict__ mu,
                  const float* __restrict__ sg, const float* __restrict__ wt,
                  const float* __restrict__ tv, float* __restrict__ ws,
                  int B, int K) {
    float acc_bce = 0.0f, acc_ll = 0.0f, acc_np = 0.0f;

    const int tid = blockIdx.x * blockDim.x + threadIdx.x;
    const int nthreads = gridDim.x * blockDim.x;

    if ((B & 3) == 0) {
        // Vector path: float4 (global_load_b128) streaming loads, NT hint
        // (dataset ~= L2 size, zero reuse -> don't thrash L2).
        const int nv = B >> 2;
        const v4f* __restrict__ pP = (const v4f*)p;
        const v4f* __restrict__ pT = (const v4f*)tv;
        for (int v = tid; v < nv; v += nthreads) {
            const v4f pp = __builtin_nontemporal_load(pP + v);
            const v4f tt = __builtin_nontemporal_load(pT + v);

            v4f lik = {0.0f, 0.0f, 0.0f, 0.0f};
            for (int k = 0; k < K; ++k) {
                const long long row = (long long)k * (long long)B;
                const v4f m = __builtin_nontemporal_load((const v4f*)(mu + row) + v);
                const v4f s = __builtin_nontemporal_load((const v4f*)(sg + row) + v);
                const v4f w = __builtin_nontemporal_load((const v4f*)(wt + row) + v);
#pragma unroll
                for (int c = 0; c < 4; ++c) {
                    const float inv = __builtin_amdgcn_rcpf(s[c] + ZIMOL_EPS);
                    const float cu = fast_sigmoid((tt[c] + ZIMOL_HB - m[c]) * inv);
                    const float cl = fast_sigmoid((tt[c] - ZIMOL_HB - m[c]) * inv);
                    lik[c] += w[c] * fmaxf(cu - cl, ZIMOL_EPS);
                }
            }
#pragma unroll
            for (int c = 0; c < 4; ++c) {
                const bool paid = tt[c] > 0.0f;
                const float logp   = fmaxf(__logf(pp[c]), -100.0f);
                const float log1mp = fmaxf(__logf(1.0f - pp[c]), -100.0f);
                acc_bce += paid ? logp : log1mp;
                const float ll = __logf(lik[c] + ZIMOL_EPS);
                acc_ll += paid ? ll : 0.0f;
                acc_np += paid ? 1.0f : 0.0f;
            }
        }
    } else {
        // Scalar fallback for non-multiple-of-4 B.
        for (int i = tid; i < B; i += nthreads) {
            zimol_element(p[i], tv[i], mu, sg, wt, (long long)i, (long long)B, K,
                          acc_bce, acc_ll, acc_np);
        }
    }

    // ---- deterministic in-block reduction: wave32 shuffle tree, then LDS ----
    const int lane = threadIdx.x & 31;
    const int wave = threadIdx.x >> 5;
#pragma unroll
    for (int off = 16; off >= 1; off >>= 1) {
        acc_bce += __shfl_xor(acc_bce, off, 32);
        acc_ll  += __shfl_xor(acc_ll,  off, 32);
        acc_np  += __shfl_xor(acc_np,  off, 32);
    }

    __shared__ float sred[3][ZIMOL_THREADS / 32];
    if (lane == 0) {
        sred[0][wave] = acc_bce;
        sred[1][wave] = acc_ll;
        sred[2][wave] = acc_np;
    }
    __syncthreads();

    if (threadIdx.x == 0) {
        float sb = 0.0f, sl = 0.0f, sn = 0.0f;
#pragma unroll
        for (int w = 0; w < ZIMOL_THREADS / 32; ++w) {
            sb += sred[0][w];
            sl += sred[1][w];
            sn += sred[2][w];
        }
        // Each block owns its own slot: deterministic, no atomics, no ws init.
        ws[blockIdx.x]                     = sb;
        ws[ZIMOL_NBLOCKS + blockIdx.x]     = sl;
        ws[2 * ZIMOL_NBLOCKS + blockIdx.x] = sn;
    }
}

// ---- finalize: single wave, matrix-unit reduction of 3 x NBLOCKS partials ----
// A = ones(16x4); D[m][n] += sum_k B[k][n].  All rows of D are identical, and
// D VGPR0 holds row0 on lanes 0-15 and row8 on lanes 16-31, so
// sum_over_32_lanes(c[0]) == 2 * total.  Exact f32 tree sum, deterministic.
__global__ void __launch_bounds__(32)
zimol_finalize_kernel(const float* __restrict__ ws, float* __restrict__ out,
                      int nblocks, float invB) {
    const int lane = threadIdx.x;   // exactly one full wave32, EXEC all ones

    v8f c0 = {}, c1 = {}, c2 = {};
    v2f ones;
    ones.x = 1.0f;
    ones.y = 1.0f;

    const float* __restrict__ pb = ws;
    const float* __restrict__ pl = ws + nblocks;
    const float* __restrict__ pn = ws + 2 * nblocks;

    for (int base = 0; base < nblocks; base += 64) {
        v2f b0, b1, b2;
        b0.x = pb[base + lane];
        b0.y = pb[base + 32 + lane];
        b1.x = pl[base + lane];
        b1.y = pl[base + 32 + lane];
        b2.x = pn[base + lane];
        b2.y = pn[base + 32 + lane];
        c0 = __builtin_amdgcn_wmma_f32_16x16x4_f32(false, ones, false, b0,
                                                   (short)0, c0, false, false);
        c1 = __builtin_amdgcn_wmma_f32_16x16x4_f32(false, ones, false, b1,
                                                   (short)0, c1, false, false);
        c2 = __builtin_amdgcn_wmma_f32_16x16x4_f32(false, ones, false, b2,
                                                   (short)0, c2, false, false);
    }

    float sb = c0[0], sl = c1[0], sn = c2[0];
#pragma unroll
    for (int off = 16; off >= 1; off >>= 1) {
        sb += __shfl_xor(sb, off, 32);
        sl += __shfl_xor(sl, off, 32);
        sn += __shfl_xor(sn, off, 32);
    }
    sb *= 0.5f;  // row0 + row8 were both summed
    sl *= 0.5f;
    sn *= 0.5f;

    if (lane == 0) {
        const float purchase_loss = -sb * invB;
        const float ltv_loss = (sn > 0.0f) ? -(sl / fmaxf(sn, 1.0f)) : 0.0f;
        out[0] = purchase_loss + ltv_loss;
    }
}

extern "C" void kernel_launch(void* const* d_in, const int* in_sizes, int n_in,
                              void* d_out, int out_size, void* d_ws, size_t ws_size,
                              hipStream_t stream) {
    const float* p  = (const float*)d_in[0];  // predicted_purchase_prob (B,)
    const float* mu = (const float*)d_in[1];  // (K, B)
    const float* sg = (const float*)d_in[2];  // (K, B)
    const float* wt = (const float*)d_in[3];  // (K, B)
    const float* tv = (const float*)d_in[4];  // true_values (B,)

    const int B = in_sizes[0];
    const int K = (B > 0) ? (in_sizes[1] / B) : 0;

    float* ws = (float*)d_ws;  // layout: [NBLOCKS bce | NBLOCKS ll | NBLOCKS npaid]

    zimol_main_kernel<<<ZIMOL_NBLOCKS, ZIMOL_THREADS, 0, stream>>>(
        p, mu, sg, wt, tv, ws, B, K);
    zimol_finalize_kernel<<<1, 32, 0, stream>>>(
        ws, (float*)d_out, ZIMOL_NBLOCKS, (B > 0) ? (1.0f / (float)B) : 0.0f);
}